// IntGELU_31688268710293
// MI455X (gfx1250) — compile-verified
//
#include <hip/hip_runtime.h>
#include <cstdint>

#define H     3072
#define TPB   256
#define NWAVE (TPB / 32)
#define EPT   (H / (TPB * 4))   // 3 float4 chunks per thread

typedef float        f4    __attribute__((ext_vector_type(4)));
typedef unsigned int u32x4 __attribute__((ext_vector_type(4)));
typedef int          i32x4 __attribute__((ext_vector_type(4)));
typedef int          i32x8 __attribute__((ext_vector_type(8)));

// int_exp_shift with reference rounding: mul-then-sub kept un-fused,
// exp2(23-q) multiply replaced by exact ldexp exponent shift.
__device__ __forceinline__ float int_exp_shift_f(float x, float x0, float lo) {
    x = __fsub_rn(__fadd_rn(x, floorf(x * 0.5f)), floorf(x * 0.0625f));
    x = fmaxf(x, lo);                          // lo = 23 * x0_int
    float q = floorf(x / x0);                  // q in [0, 23]
    float r = __fsub_rn(x, __fmul_rn(x0, q));  // x - x0*q, two roundings (as ref)
    float e = __fsub_rn(r * 0.5f, x0);         // r*0.5 exact -> same as ref
    e = floorf(ldexpf(e, 23 - (int)q));        // e * 2^(23-q), exact scale
    return fmaxf(e, 0.0f);
}

__global__ void __launch_bounds__(TPB)
intgelu_kernel(const float* __restrict__ x, const float* __restrict__ sfp,
               float* __restrict__ out, int nrows) {
    __shared__ __align__(16) float buf[H];     // one 12KB row staged in LDS
    __shared__ float redmx[NWAVE];
    __shared__ float redmn[NWAVE];

    const int row = blockIdx.x;
    const int t   = threadIdx.x;
    const float sf = sfp[0];

    const float* rowp = x + (size_t)row * H;

    // ---- CDNA5 Tensor Data Mover: one D# fetches the whole 12KB row ----
    if (t < 32) {   // TDM issue is per-wave: wave 0 only (EXEC is ignored by TDM)
        uint64_t gaddr = (uint64_t)(uintptr_t)rowp;
        uint32_t ldsa  = (uint32_t)(uintptr_t)(&buf[0]);  // LDS byte offset

        u32x4 g0;
        g0.x = 1u;                                  // count=1 valid descriptor
        g0.y = ldsa;                                // lds_addr      [63:32]
        g0.z = (uint32_t)gaddr;                     // global_addr   [95:64]
        g0.w = (uint32_t)((gaddr >> 32) & 0x1FFFFFFu) | (2u << 30); // [120:96] | type=2

        i32x8 g1;
        g1[0] = (int)(2u << 16);                    // data_size=2 (4B); mask/pad/iter=0
        g1[1] = (int)(((uint32_t)H & 0xFFFFu) << 16);   // tensor_dim0[15:0] @ [63:48]
        g1[2] = (int)(1u << 16);                    // tensor_dim0[31:16]=0, tensor_dim1=1
        g1[3] = (int)(((uint32_t)H & 0xFFFFu) << 16);   // tile_dim0=3072 @ [127:112]
        g1[4] = 1;                                  // tile_dim1=1, tile_dim2=0
        g1[5] = H;                                  // tensor_dim0_stride[31:0]
        g1[6] = 0;                                  // stride hi / dim1_stride lo
        g1[7] = 0;

        i32x4 gz = {0, 0, 0, 0};                    // groups 2/3: >2D dims unused
        i32x8 z8 = {0, 0, 0, 0, 0, 0, 0, 0};        // extra group (clang-23 6-arg form)

        __builtin_amdgcn_tensor_load_to_lds(g0, g1, gz, gz, z8, 0);
        __builtin_amdgcn_s_wait_tensorcnt(0);
    }
    __syncthreads();   // publish TDM-written LDS row to all 8 waves

    // ---- pass 1: raw min/max (division by sf is monotonic -> commutes) ----
    float mx = -INFINITY, mn = INFINITY;
#pragma unroll
    for (int k = 0; k < EPT; ++k) {
        f4 v = ((const f4*)buf)[k * TPB + t];
        mx = fmaxf(mx, fmaxf(fmaxf(v.x, v.y), fmaxf(v.z, v.w)));
        mn = fminf(mn, fminf(fminf(v.x, v.y), fminf(v.z, v.w)));
    }
#pragma unroll
    for (int m = 16; m >= 1; m >>= 1) {        // wave32 butterfly
        mx = fmaxf(mx, __shfl_xor(mx, m, 32));
        mn = fminf(mn, __shfl_xor(mn, m, 32));
    }
    if ((t & 31) == 0) { redmx[t >> 5] = mx; redmn[t >> 5] = mn; }
    __syncthreads();
    mx = redmx[0]; mn = redmn[0];
#pragma unroll
    for (int j = 1; j < NWAVE; ++j) {
        mx = fmaxf(mx, redmx[j]);
        mn = fminf(mn, redmn[j]);
    }

    // max(x/sf) == max(x)/sf for sf>0 (min(x)/sf if sf<0), bitwise
    const float rowMax = (sf > 0.0f) ? (mx / sf) : (mn / sf);
    const float x0     = floorf(-1.0f / (sf * 1.702f));
    const float lo     = 23.0f * x0;
    const float eMax   = int_exp_shift_f(-rowMax, x0, lo);
    const float out_sf = sf * (1.0f / 128.0f);

    // ---- pass 2: one IEEE division per element, NT b128 stores ----
    f4* orow = (f4*)(out + (size_t)row * H);
#pragma unroll
    for (int k = 0; k < EPT; ++k) {
        int idx4 = k * TPB + t;
        f4 v = ((const f4*)buf)[idx4];
        f4 o;
#pragma unroll
        for (int c = 0; c < 4; ++c) {
            float p  = v[c] / sf;                         // pre_x_int
            float e  = int_exp_shift_f(p - rowMax, x0, lo);
            float es = fminf(e + eMax, 2147483647.0f);
            float factor = floorf(2147483647.0f / es);
            float sig = floorf(__fmul_rn(e, factor) * (1.0f / 16777216.0f));
            o[c] = __fmul_rn(__fmul_rn(p, sig), out_sf);
        }
        __builtin_nontemporal_store(o, &orow[idx4]);
    }

    // second tuple output: scalar out_sf appended after the tensor
    if (row == 0 && t == 0) out[(size_t)nrows * H] = out_sf;
}

extern "C" void kernel_launch(void* const* d_in, const int* in_sizes, int n_in,
                              void* d_out, int out_size, void* d_ws, size_t ws_size,
                              hipStream_t stream) {
    const float* x  = (const float*)d_in[0];
    const float* sf = (const float*)d_in[1];
    float* out = (float*)d_out;
    int nrows = in_sizes[0] / H;   // 64*196 = 12544 rows
    intgelu_kernel<<<nrows, TPB, 0, stream>>>(x, sf, out, nrows);
}